// BahdanauAttention_30786325578155
// MI455X (gfx1250) — compile-verified
//
#include <hip/hip_runtime.h>
#include <hip/hip_bf16.h>
#include <math.h>

// ---------------------------------------------------------------------------
// Bahdanau attention, fused for MI455X (gfx1250, wave32, WMMA).
//   B=64, N=2048, H=D=A=512  (fp32 end-to-end; fp32 WMMA => exact math)
// ---------------------------------------------------------------------------

typedef __attribute__((ext_vector_type(2))) float v2f;
typedef __attribute__((ext_vector_type(8))) float v8f;

static constexpr int Bb   = 64;
static constexpr int Nn   = 2048;
static constexpr int Hh   = 512;
static constexpr int Dd   = 512;
static constexpr int Aa   = 512;
static constexpr int LDSROW = 516;        // 512 + 4-dword pad: bank-conflict-free
static constexpr float NEG_INF = -1.0e9f;

// Hardware V_TANH_F32 if the toolchain exposes it (gfx1250 TRANS op),
// else precise OCML tanhf.
#if defined(__has_builtin)
#if __has_builtin(__builtin_amdgcn_tanhf)
#define FAST_TANH(x) __builtin_amdgcn_tanhf(x)
#endif
#endif
#ifndef FAST_TANH
#define FAST_TANH(x) tanhf(x)
#endif

// --------------------------- proj_h = h @ W_h^T ----------------------------
__global__ __launch_bounds__(256) void proj_h_kernel(
    const float* __restrict__ h, const float* __restrict__ W_h,
    float* __restrict__ proj_h)
{
    int idx = blockIdx.x * 256 + threadIdx.x;       // 64*512 = 32768
    int b = idx >> 9;
    int a = idx & 511;
    const float4* hr = (const float4*)(h + (size_t)b * Hh);
    const float4* wr = (const float4*)(W_h + (size_t)a * Hh);
    float acc = 0.0f;
#pragma unroll 4
    for (int k = 0; k < Hh / 4; ++k) {
        float4 x = hr[k], w = wr[k];
        acc += x.x * w.x + x.y * w.y + x.z * w.z + x.w * w.w;
    }
    proj_h[idx] = acc;
}

// ------------- fused: proj_e (WMMA f32) -> tanh -> dot(v) -> scores --------
// grid: (B, N/16); block: 256 threads = 8 waves.
// wave w computes A-columns [w*64, w*64+64) as 4 16x16 WMMA tiles over K=512.
__global__ __launch_bounds__(256) void energy_scores_kernel(
    const float* __restrict__ enc,     // (B,N,512)
    const float* __restrict__ W_e,     // (512,512) row-major (a, d)
    const float* __restrict__ proj_h,  // (B,512)
    const float* __restrict__ v,       // (512)
    float* __restrict__ scores)        // (B,N)
{
    __shared__ float lds_enc[16 * LDSROW];
    __shared__ float swave[8][16];

    const int b   = blockIdx.x;
    const int m0  = blockIdx.y * 16;
    const int tid = threadIdx.x;
    const int wid  = tid >> 5;
    const int lane = tid & 31;

    // ---- cooperative load: 16x512 enc tile -> padded LDS (float4 loads, scalar LDS stores) ----
    const float* src = enc + ((size_t)b * Nn + m0) * Dd;
    for (int i = tid; i < 16 * 128; i += 256) {          // 2048 float4's
        int r  = i >> 7;
        int c4 = i & 127;
        float4 val = *(const float4*)(src + (size_t)r * Dd + c4 * 4);
        float* dst = &lds_enc[r * LDSROW + c4 * 4];
        dst[0] = val.x; dst[1] = val.y; dst[2] = val.z; dst[3] = val.w;
    }
    __syncthreads();

    // ---- per-lane fragment addressing (ISA 16x4 / 4x16 fp32 layouts) ----
    const int mrow  = lane & 15;     // M (A-frag) or N (B-frag) index
    const int khalf = lane >> 4;     // 0: K={0,1}   1: K={2,3}
    const int a_base = wid * 64;

    const float* arow = &lds_enc[mrow * LDSROW + 2 * khalf];
    const float* we0 = W_e + (size_t)(a_base +  0 + mrow) * Dd + 2 * khalf;
    const float* we1 = W_e + (size_t)(a_base + 16 + mrow) * Dd + 2 * khalf;
    const float* we2 = W_e + (size_t)(a_base + 32 + mrow) * Dd + 2 * khalf;
    const float* we3 = W_e + (size_t)(a_base + 48 + mrow) * Dd + 2 * khalf;

    v8f acc0 = {}, acc1 = {}, acc2 = {}, acc3 = {};

#pragma unroll 4
    for (int k = 0; k < Dd; k += 4) {
        v2f af = *(const v2f*)(arow + k);
        v2f b0 = *(const v2f*)(we0 + k);
        v2f b1 = *(const v2f*)(we1 + k);
        v2f b2 = *(const v2f*)(we2 + k);
        v2f b3 = *(const v2f*)(we3 + k);
        acc0 = __builtin_amdgcn_wmma_f32_16x16x4_f32(false, af, false, b0,
                                                     (short)0, acc0, false, false);
        acc1 = __builtin_amdgcn_wmma_f32_16x16x4_f32(false, af, false, b1,
                                                     (short)0, acc1, false, false);
        acc2 = __builtin_amdgcn_wmma_f32_16x16x4_f32(false, af, false, b2,
                                                     (short)0, acc2, false, false);
        acc3 = __builtin_amdgcn_wmma_f32_16x16x4_f32(false, af, false, b3,
                                                     (short)0, acc3, false, false);
    }

    // ---- epilogue: tanh(proj_e + proj_h) * v, reduce over A in-register ----
    // C/D layout: lane holds N = a0 + (lane&15); VGPR j holds M = m0 + j + 8*(lane>>4)
    float partial[8];
#pragma unroll
    for (int j = 0; j < 8; ++j) partial[j] = 0.0f;

    const float* phb = proj_h + (size_t)b * Aa;
#pragma unroll
    for (int ct = 0; ct < 4; ++ct) {
        int n    = a_base + ct * 16 + mrow;
        float ph = phb[n];
        float vv = v[n];
        v8f c = (ct == 0) ? acc0 : (ct == 1) ? acc1 : (ct == 2) ? acc2 : acc3;
#pragma unroll
        for (int j = 0; j < 8; ++j)
            partial[j] += vv * FAST_TANH(c[j] + ph);
    }

    // reduce across the 16 lanes of each half-wave (N dimension)
#pragma unroll
    for (int j = 0; j < 8; ++j) {
        float p = partial[j];
        p += __shfl_xor(p, 1);
        p += __shfl_xor(p, 2);
        p += __shfl_xor(p, 4);
        p += __shfl_xor(p, 8);
        partial[j] = p;
    }
    if ((lane & 15) == 0) {
        int rbase = khalf * 8;                 // lane0 -> rows 0..7, lane16 -> 8..15
#pragma unroll
        for (int j = 0; j < 8; ++j) swave[wid][rbase + j] = partial[j];
    }
    __syncthreads();

    if (tid < 16) {
        float s = 0.0f;
#pragma unroll
        for (int w = 0; w < 8; ++w) s += swave[w][tid];
        scores[(size_t)b * Nn + m0 + tid] = s;
    }
}

// ------------------------- masked softmax over N ---------------------------
__global__ __launch_bounds__(256) void softmax_kernel(
    const float* __restrict__ scores, const int* __restrict__ mask,
    float* __restrict__ alpha)
{
    __shared__ float red[256];
    const int b = blockIdx.x;
    const int tid = threadIdx.x;

    float m = -3.0e38f;
    for (int n = tid; n < Nn; n += 256) {
        float s = (mask[(size_t)b * Nn + n] == 0) ? NEG_INF
                                                  : scores[(size_t)b * Nn + n];
        m = fmaxf(m, s);
    }
    red[tid] = m; __syncthreads();
    for (int s = 128; s > 0; s >>= 1) {
        if (tid < s) red[tid] = fmaxf(red[tid], red[tid + s]);
        __syncthreads();
    }
    float mx = red[0];
    __syncthreads();

    float sum = 0.0f;
    for (int n = tid; n < Nn; n += 256) {
        float s = (mask[(size_t)b * Nn + n] == 0) ? NEG_INF
                                                  : scores[(size_t)b * Nn + n];
        float e = expf(s - mx);
        alpha[(size_t)b * Nn + n] = e;
        sum += e;
    }
    red[tid] = sum; __syncthreads();
    for (int s = 128; s > 0; s >>= 1) {
        if (tid < s) red[tid] += red[tid + s];
        __syncthreads();
    }
    float inv = 1.0f / red[0];
    for (int n = tid; n < Nn; n += 256)
        alpha[(size_t)b * Nn + n] *= inv;
}

// ---------------- context = alpha @ enc_mem  (2-stage, no atomics) ---------
__global__ __launch_bounds__(256) void context_partial_kernel(
    const float* __restrict__ alpha, const float* __restrict__ enc,
    float* __restrict__ partials)               // (B,16,512)
{
    const int b = blockIdx.x;
    const int chunk = blockIdx.y;               // 16 chunks of 128 n's
    const int d = threadIdx.x;                  // covers d and d+256
    const float* al = alpha + (size_t)b * Nn + chunk * 128;
    const float* e  = enc + ((size_t)b * Nn + (size_t)chunk * 128) * Dd;
    float acc0 = 0.0f, acc1 = 0.0f;
#pragma unroll 4
    for (int n = 0; n < 128; ++n) {
        float a = al[n];
        acc0 += a * e[(size_t)n * Dd + d];
        acc1 += a * e[(size_t)n * Dd + d + 256];
    }
    float* p = partials + ((size_t)b * 16 + chunk) * Dd;
    p[d] = acc0;
    p[d + 256] = acc1;
}

__global__ __launch_bounds__(256) void context_reduce_kernel(
    const float* __restrict__ partials, float* __restrict__ ctx)
{
    int idx = blockIdx.x * 256 + threadIdx.x;   // 64*512
    int b = idx >> 9;
    int d = idx & 511;
    float s = 0.0f;
#pragma unroll
    for (int c = 0; c < 16; ++c)
        s += partials[((size_t)b * 16 + c) * Dd + d];
    ctx[idx] = s;
}

// ---------------------------------------------------------------------------
extern "C" void kernel_launch(void* const* d_in, const int* in_sizes, int n_in,
                              void* d_out, int out_size, void* d_ws, size_t ws_size,
                              hipStream_t stream)
{
    const float* h     = (const float*)d_in[0];   // (B,H)
    const float* enc   = (const float*)d_in[1];   // (B,N,D)
    const int*   mask  = (const int*)  d_in[2];   // (B,N)
    const float* W_h   = (const float*)d_in[3];   // (A,H)
    const float* W_e   = (const float*)d_in[4];   // (A,D)
    const float* v     = (const float*)d_in[5];   // (A)

    float* out     = (float*)d_out;
    float* context = out;                 // (B,D)   = 32768 floats
    float* alpha   = out + Bb * Dd;       // (B,N)   = 131072 floats

    float* ws       = (float*)d_ws;
    float* proj_h   = ws;                                  // 32768
    float* scores   = ws + Bb * Aa;                        // 131072
    float* partials = ws + Bb * Aa + Bb * Nn;              // 64*16*512 = 524288

    proj_h_kernel<<<dim3((Bb * Aa) / 256), dim3(256), 0, stream>>>(h, W_h, proj_h);

    energy_scores_kernel<<<dim3(Bb, Nn / 16), dim3(256), 0, stream>>>(
        enc, W_e, proj_h, v, scores);

    softmax_kernel<<<dim3(Bb), dim3(256), 0, stream>>>(scores, mask, alpha);

    context_partial_kernel<<<dim3(Bb, 16), dim3(256), 0, stream>>>(
        alpha, enc, partials);

    context_reduce_kernel<<<dim3((Bb * Dd) / 256), dim3(256), 0, stream>>>(
        partials, context);
}